// RuleAwareProjection_24034636988908
// MI455X (gfx1250) — compile-verified
//
#include <hip/hip_runtime.h>
#include <hip/hip_bf16.h>

typedef float v2f __attribute__((ext_vector_type(2)));
typedef float v8f __attribute__((ext_vector_type(8)));
typedef unsigned int u32x4 __attribute__((ext_vector_type(4)));
typedef int i32x4 __attribute__((ext_vector_type(4)));
typedef int i32x8 __attribute__((ext_vector_type(8)));

#define NTOK    16384
#define INF     2048
#define OUTF    2048
#define SR      45      // shared rank
#define SRP     48      // padded rank (multiple of 4 for WMMA K-steps)
#define HSTR    52      // LDS row stride for h (bank-conflict padding)
#define NB      32      // blocks
#define BS      64      // block size
#define RANK    8
#define NRULES  256
#define TILE_M  16      // tokens per workgroup

// ---- x-tile staging via TDM (per-wave 16 rows x 256 cols, 2 pad DWORDs / 128) ----
#define XT_STRIDE_F 260                    // floats per staged row (256 + 2*2 pad)
#define XT_WAVE_F   (TILE_M * XT_STRIDE_F) // 4160 floats per wave region
#define XT_WAVE_B   (XT_WAVE_F * 4)        // 16640 bytes

// ---- LDS layout (bytes) ----
#define OFF_XT   0                         // 8 * 16640 = 133120 (phase A only)
#define OFF_AD   0                         // float[16][1024] = 65536 (overlays XT, phase C3/B)
#define OFF_H    65536                     // float[16][HSTR] = 3328  (overlays XT, post-barrier)
#define OFF_R2   133120                    // overlay: HP (phase A) vs HID+SEL (phase C)
#define OFF_HP   OFF_R2                    // float[8][16*48] = 24576
#define OFF_HID  OFF_R2                    // float[16][32*8] = 16384
#define OFF_SEL  (OFF_R2 + 16384)          // float[16][32]   = 2048
#define OFF_RID  (OFF_R2 + 24576)          // int[16]         = 64
#define SMEM_BYTES (OFF_RID + 64)          // 157760

static __device__ __forceinline__ v8f wmma_f32(v2f a, v2f b, v8f c) {
    // D = A(16x4,f32) * B(4x16,f32) + C(16x16,f32)
    return __builtin_amdgcn_wmma_f32_16x16x4_f32(
        /*neg_a=*/false, a, /*neg_b=*/false, b,
        /*c_mod=*/(short)0, c, /*reuse_a=*/false, /*reuse_b=*/false);
}

// ---------------------------------------------------------------------------
// Pre-pass: per-RULE block-selector softmax (sel depends only on the rule id)
// ---------------------------------------------------------------------------
__global__ void selw_kernel(const float* __restrict__ logits, float* __restrict__ selw) {
    const int r = blockIdx.x * blockDim.x + threadIdx.x;
    if (r >= NRULES) return;
    const float invt = 5.65685424949238f;  // sqrt(32)
    float v[NB];
    float mx = -3.402823466e38f;
    for (int b = 0; b < NB; ++b) {
        float t = logits[r * NB + b] * invt;
        v[b] = t;
        mx = fmaxf(mx, t);
    }
    float s = 0.f;
    for (int b = 0; b < NB; ++b) { float e = __expf(v[b] - mx); v[b] = e; s += e; }
    const float inv = 1.0f / s;
    for (int b = 0; b < NB; ++b) selw[r * NB + b] = v[b] * inv;
}

// ---------------------------------------------------------------------------
// Fused main kernel: 16 tokens per workgroup, 8 waves
// ---------------------------------------------------------------------------
__global__ __launch_bounds__(256)
void rule_proj_kernel(const float* __restrict__ x,
                      const int*   __restrict__ rule_ids,
                      const float* __restrict__ shared_in,   // [2048][45]
                      const float* __restrict__ shared_out,  // [45][2048]
                      const float* __restrict__ rule_in,     // [256][64][8]
                      const float* __restrict__ rule_out,    // [256][8][64]
                      const float* __restrict__ selw,        // [256][32]
                      float*       __restrict__ out)
{
    __shared__ __align__(16) char smem[SMEM_BYTES];
    float* AD  = (float*)(smem + OFF_AD);
    float* H   = (float*)(smem + OFF_H);
    float* HP  = (float*)(smem + OFF_HP);
    float* HID = (float*)(smem + OFF_HID);
    float* SEL = (float*)(smem + OFF_SEL);
    int*   RID = (int*)  (smem + OFF_RID);

    const int tok0 = blockIdx.x * TILE_M;
    const int tid  = threadIdx.x;
    const int w    = tid >> 5;      // wave id 0..7
    const int lane = tid & 31;
    const int lr   = lane & 15;     // row/col within a 16-wide tile
    const int hi   = lane >> 4;     // lane-half: selects K pair / M offset 8

    // ---------- TDM: async-stage this wave's K-chunk of the x tile into LDS ----------
    // 2D tile: 16 rows (tokens) x 256 cols, global row stride 2048 floats,
    // LDS padding of 2 DWORDs every 128 DWORDs -> staged row stride 260 floats
    // (bank = 4*lane -> conflict-free WMMA-A feeds).
    {
        const unsigned long long ga = (unsigned long long)(uintptr_t)x
            + (((unsigned long long)tok0 * INF + (unsigned)(w * 256)) << 2);
        u32x4 g0;
        g0[0] = 1u;                                    // count = 1 (valid descriptor)
        g0[1] = (unsigned)(OFF_XT + w * XT_WAVE_B);    // lds_addr (bytes)
        g0[2] = (unsigned)(ga & 0xFFFFFFFFu);          // global_addr[31:0]
        g0[3] = ((unsigned)(ga >> 32) & 0x01FFFFFFu) | (2u << 30);  // addr[56:32] | type=2
        i32x8 g1;
        g1[0] = (2 << 16)      // data_size = 4 bytes
              | (1 << 20)      // pad_enable
              | (6 << 22)      // pad_interval code 6 = 128 DWORDs
              | (1 << 25);     // pad_amount  code 1 = 2 DWORDs
        g1[1] = (int)(2048u << 16);   // tensor_dim0[15:0] = 2048
        g1[2] = (int)(16u << 16);     // tensor_dim1[15:0] = 16
        g1[3] = (int)(256u << 16);    // tile_dim0 = 256
        g1[4] = 16;                   // tile_dim1 = 16 rows
        g1[5] = 2048;                 // tensor_dim0_stride[31:0] = 2048
        g1[6] = 0;
        g1[7] = 0;
        i32x4 gz  = {0, 0, 0, 0};             // groups 2/3 unused for a 2D tensor
        i32x8 gz8 = {0, 0, 0, 0, 0, 0, 0, 0}; // extra group (6-arg toolchain form)
        __builtin_amdgcn_tensor_load_to_lds(g0, g1, gz, gz, gz8, /*cpol=*/0);
    }

    if (tid < TILE_M) RID[tid] = rule_ids[tok0 + tid];

    // Each wave consumes only its own staged region -> wait on own TENSORcnt only.
    __builtin_amdgcn_s_wait_tensorcnt(0);
    asm volatile("" ::: "memory");

    // ---------- Phase A: h = x_tile @ shared_in, K=2048 split over 8 waves ----------
    v8f acc0 = {}, acc1 = {}, acc2 = {};
    {
        const float* XTw = (const float*)(smem + OFF_XT) + w * XT_WAVE_F + lr * XT_STRIDE_F;
        const int c0 = lr, c1 = 16 + lr, c2 = 32 + lr;
        const int c2c = (c2 < SR) ? c2 : (SR - 1);  // safe addr; cols>=45 zeroed at reduce
        const int kbase = w * (INF / 8);
        for (int s = 0; s < (INF / 8) / 4; ++s) {
            const int j  = 4 * s + 2 * hi;          // col within staged chunk
            const int jj = j + ((j >> 7) << 1);     // + LDS pad correction
            v2f a = *(const v2f*)(XTw + jj);        // A[lr][kk], A[lr][kk+1] from LDS
            const int kk = kbase + j;
            const float* b0 = shared_in + (size_t)kk * SR;
            const float* b1 = b0 + SR;
            v2f b;
            b.x = b0[c0]; b.y = b1[c0];
            acc0 = wmma_f32(a, b, acc0);
            b.x = b0[c1]; b.y = b1[c1];
            acc1 = wmma_f32(a, b, acc1);
            b.x = b0[c2c]; b.y = b1[c2c];
            acc2 = wmma_f32(a, b, acc2);
        }
        float* hp = HP + w * (TILE_M * SRP);
#pragma unroll
        for (int v = 0; v < 8; ++v) {
            const int m = v + 8 * hi;
            hp[m * SRP + c0] = acc0[v];
            hp[m * SRP + c1] = acc1[v];
            hp[m * SRP + c2] = acc2[v];
        }
    }
    __syncthreads();

    // cross-wave reduction of partial h; pad columns 45..47 with exact zeros
    // (XT region is dead from here on: H/AD overlay it)
    for (int e = tid; e < TILE_M * SRP; e += 256) {
        const int m = e / SRP, c = e % SRP;
        float sum = 0.f;
        if (c < SR) {
#pragma unroll
            for (int ww = 0; ww < 8; ++ww) sum += HP[ww * (TILE_M * SRP) + e];
        }
        H[m * HSTR + c] = sum;
    }
    __syncthreads();

    // ---------- Phase C1: gather per-token selector weights (HP is dead now) ----------
    for (int i = tid; i < TILE_M * NB; i += 256) {
        const int m = i >> 5, b = i & (NB - 1);
        SEL[m * NB + b] = selw[RID[m] * NB + b];
    }

    // ---------- Phase C2: hidden[t][32x8] = x_blocks[t] @ rule_in[rid[t]] ----------
    for (int tt = 0; tt < 2; ++tt) {
        const int t    = 2 * w + tt;
        const int rule = RID[t];
        const float* xr  = x + (size_t)(tok0 + t) * INF;
        const float* rin = rule_in + (size_t)rule * (BS * RANK);  // [64][8]
        const int ncl = lr & (RANK - 1);   // clamp B col; cols 8..15 never stored
#pragma unroll
        for (int mt = 0; mt < 2; ++mt) {   // block rows 0..15 / 16..31
            v8f ch = {};
            const float* xa = xr + (mt * 16 + lr) * BS;
#pragma unroll
            for (int s = 0; s < BS / 4; ++s) {
                const int kk = 4 * s + 2 * hi;
                v2f a = *(const v2f*)(xa + kk);
                v2f b;
                b.x = rin[kk * RANK + ncl];
                b.y = rin[(kk + 1) * RANK + ncl];
                ch = wmma_f32(a, b, ch);
            }
            if (lr < RANK) {
#pragma unroll
                for (int v = 0; v < 8; ++v) {
                    const int m = v + 8 * hi;
                    HID[t * (NB * RANK) + (mt * 16 + m) * RANK + lr] = ch[v];
                }
            }
        }
    }
    __syncthreads();

    // ---------- Phases C3 + B, in two column halves (AD holds 16x1024) ----------
    for (int h = 0; h < 2; ++h) {
        // C3: weighted adapter for blocks [h*16, h*16+16) into AD
        for (int tt = 0; tt < 2; ++tt) {
            const int t    = 2 * w + tt;
            const int rule = RID[t];
            const float* rout = rule_out + (size_t)rule * (RANK * BS);  // [8][64]
            const float* hidt = HID + t * (NB * RANK);
#pragma unroll
            for (int nt = 0; nt < 4; ++nt) {
                v8f ca = {};
#pragma unroll
                for (int s = 0; s < 2; ++s) {          // K = 8
                    const int kk = 4 * s + 2 * hi;
                    v2f a;
                    a.x = hidt[(h * 16 + lr) * RANK + kk];
                    a.y = hidt[(h * 16 + lr) * RANK + kk + 1];
                    v2f b;
                    b.x = rout[kk * BS + nt * 16 + lr];
                    b.y = rout[(kk + 1) * BS + nt * 16 + lr];
                    ca = wmma_f32(a, b, ca);
                }
#pragma unroll
                for (int v = 0; v < 8; ++v) {
                    const int bl = v + 8 * hi;                         // local block row
                    AD[t * (OUTF / 2) + bl * BS + nt * 16 + lr] =
                        ca[v] * SEL[t * NB + h * 16 + bl];
                }
            }
        }
        __syncthreads();

        // B: out = adapter + h @ shared_out for columns [h*1024, h*1024+1024)
        for (int i = 0; i < 8; ++i) {
            const int nt  = h * 64 + w * 8 + i;
            const int col = nt * 16 + lr;
            const int lcol = col - h * (OUTF / 2);
            v8f c;
#pragma unroll
            for (int v = 0; v < 8; ++v) c[v] = AD[(v + 8 * hi) * (OUTF / 2) + lcol];
#pragma unroll
            for (int s = 0; s < SRP / 4; ++s) {        // K = 48 (padded)
                const int kk = 4 * s + 2 * hi;
                v2f a;
                a.x = H[lr * HSTR + kk];               // zero for kk >= 45
                a.y = H[lr * HSTR + kk + 1];
                const int r0 = (kk     < SR) ? kk     : SR - 1;  // safe addr; A==0 there
                const int r1 = (kk + 1 < SR) ? kk + 1 : SR - 1;
                v2f b;
                b.x = shared_out[(size_t)r0 * OUTF + col];
                b.y = shared_out[(size_t)r1 * OUTF + col];
                c = wmma_f32(a, b, c);
            }
            float* op = out + (size_t)tok0 * OUTF + col;
#pragma unroll
            for (int v = 0; v < 8; ++v)
                __builtin_nontemporal_store(c[v], op + (size_t)(v + 8 * hi) * OUTF);
        }
        __syncthreads();
    }
}

// ---------------------------------------------------------------------------
extern "C" void kernel_launch(void* const* d_in, const int* in_sizes, int n_in,
                              void* d_out, int out_size, void* d_ws, size_t ws_size,
                              hipStream_t stream) {
    (void)in_sizes; (void)n_in; (void)out_size; (void)ws_size;
    const float* x          = (const float*)d_in[0];
    const int*   rule_ids   = (const int*)  d_in[1];
    const float* shared_in  = (const float*)d_in[2];
    const float* shared_out = (const float*)d_in[3];
    const float* rule_in    = (const float*)d_in[4];
    const float* rule_out   = (const float*)d_in[5];
    const float* logits     = (const float*)d_in[6];
    float* out  = (float*)d_out;
    float* selw = (float*)d_ws;   // 256*32 floats = 32 KB of workspace

    selw_kernel<<<1, 256, 0, stream>>>(logits, selw);
    rule_proj_kernel<<<NTOK / TILE_M, 256, 0, stream>>>(
        x, rule_ids, shared_in, shared_out, rule_in, rule_out, selw, out);
}